// BatchSparseIndexSubsetAttention_2052994367933
// MI455X (gfx1250) — compile-verified
//
#include <hip/hip_runtime.h>
#include <hip/hip_bf16.h>
#include <math.h>

typedef __attribute__((ext_vector_type(2))) float v2f;
typedef __attribute__((ext_vector_type(8))) float v8f;

#define NN 4096
#define LL 64
#define MM 256
#define HH 8
#define DD 32
#define D2C 16
#define PP 3
#define LDST 260   // 256 + 4 pad: conflict-free b64 A-frag reads and column D writes

__launch_bounds__(256)
__global__ void bsisa_kernel(
    const float* __restrict__ sparse_values,   // (NNZ, 256)
    const int*   __restrict__ linear_index,    // (N, 64)
    const int*   __restrict__ is_specified,    // (N, 64)
    const float* __restrict__ query,           // (N, 256)
    const float* __restrict__ key_weight,      // (256, 256)
    const float* __restrict__ value_weight,    // (256, 256)
    const float* __restrict__ key_positions,   // (N, 64, 3)
    const float* __restrict__ rope_freqs,      // (3, 1, 8, 16)
    float* __restrict__ out,                   // (N, 256)
    float* __restrict__ out_spec)              // (N, 64) as 0/1 float
{
  __shared__ float sG[LL * LDST];    // gathered rows (~66.5 KB)
  __shared__ float sK[LL * LDST];    // K projection  (~66.5 KB)
  __shared__ float sQ[MM];
  __shared__ float sS[HH * LL];      // scores -> attn
  __shared__ float sGB[HH * MM];     // attn @ G
  __shared__ int   sSpec[LL];
  __shared__ int   sIdx[LL];

  const int n    = blockIdx.x;
  const int t    = threadIdx.x;
  const int wave = t >> 5;
  const int lane = t & 31;

  // ---- stage flags / indices / q ----
  if (t < LL) {
    int spec = is_specified[n * LL + t];
    sSpec[t] = spec;
    sIdx[t]  = linear_index[n * LL + t];
    out_spec[(size_t)n * LL + t] = spec ? 1.0f : 0.0f;
  }
  sQ[t] = query[(size_t)n * MM + t];
  __syncthreads();

  // ---- gather 64 x 256 fp32 rows into LDS (zero where unspecified) ----
  // 64 rows * 64 float4 = 4096 float4 loads, 16 per thread; L2-resident table.
  for (int idx = t; idx < LL * 64; idx += 256) {
    int l  = idx >> 6;
    int c4 = idx & 63;
    float4 val = make_float4(0.f, 0.f, 0.f, 0.f);
    if (sSpec[l]) {
      val = *(const float4*)(sparse_values + (size_t)sIdx[l] * MM + c4 * 4);
    }
    *(float4*)&sG[l * LDST + c4 * 4] = val;
  }
  __syncthreads();

  // ---- K = G @ Wk^T via V_WMMA_F32_16X16X4_F32 ----
  // D(16x16) tiles: 4 row tiles (64 rows) x 16 col tiles (256 cols).
  // Each wave: 2 col tiles, 4 row tiles each, k-loop 256 in steps of 4.
  const int row16 = lane & 15;
  const int kb    = (lane >> 4) << 1;        // 0 or 2 (K sub-slot per half-wave)
  const int mb    = (lane >> 4) << 3;        // 0 or 8 (D row base per half-wave)

  for (int ot = wave; ot < 16; ot += 8) {
    const float* wrow = key_weight + (size_t)(ot * 16 + row16) * MM; // row o of Wk
    v8f acc0 = {}, acc1 = {}, acc2 = {}, acc3 = {};
    const float* g0 = &sG[(0 * 16 + row16) * LDST];
    const float* g1 = &sG[(1 * 16 + row16) * LDST];
    const float* g2 = &sG[(2 * 16 + row16) * LDST];
    const float* g3 = &sG[(3 * 16 + row16) * LDST];
    for (int k0 = 0; k0 < MM; k0 += 4) {
      // B frag: B[k, o] = Wk[o, k] -> 2 consecutive floats of weight row o
      v2f b  = *(const v2f*)(wrow + k0 + kb);
      // A frags: 2 consecutive floats of gathered row (LDS, bank-conflict-free)
      v2f a0 = *(const v2f*)(g0 + k0 + kb);
      v2f a1 = *(const v2f*)(g1 + k0 + kb);
      v2f a2 = *(const v2f*)(g2 + k0 + kb);
      v2f a3 = *(const v2f*)(g3 + k0 + kb);
      acc0 = __builtin_amdgcn_wmma_f32_16x16x4_f32(false, a0, false, b, (short)0, acc0, false, false);
      acc1 = __builtin_amdgcn_wmma_f32_16x16x4_f32(false, a1, false, b, (short)0, acc1, false, false);
      acc2 = __builtin_amdgcn_wmma_f32_16x16x4_f32(false, a2, false, b, (short)0, acc2, false, false);
      acc3 = __builtin_amdgcn_wmma_f32_16x16x4_f32(false, a3, false, b, (short)0, acc3, false, false);
    }
    // D layout: lane = col (N), VGPR j = row (mb + j)
    const int col = ot * 16 + row16;
#pragma unroll
    for (int j = 0; j < 8; ++j) {
      sK[(0 * 16 + mb + j) * LDST + col] = acc0[j];
      sK[(1 * 16 + mb + j) * LDST + col] = acc1[j];
      sK[(2 * 16 + mb + j) * LDST + col] = acc2[j];
      sK[(3 * 16 + mb + j) * LDST + col] = acc3[j];
    }
  }
  __syncthreads();

  // ---- RoPE in place on K: 64 l * 8 h * 16 pairs ----
  for (int idx = t; idx < LL * HH * D2C; idx += 256) {
    int l  = idx >> 7;
    int r  = idx & 127;
    int h  = r >> 4;
    int d2 = r & 15;
    const float* pos = key_positions + ((size_t)n * LL + l) * PP;
    int fo = h * 16 + d2;                 // rope_freqs[p][0][h][d2]
    float ang = pos[0] * rope_freqs[0 * 128 + fo]
              + pos[1] * rope_freqs[1 * 128 + fo]
              + pos[2] * rope_freqs[2 * 128 + fo];
    float s, c;
    __sincosf(ang, &s, &c);
    float* kp = &sK[l * LDST + h * DD + 2 * d2];
    float ke = kp[0], ko = kp[1];
    kp[0] = ke * c - ko * s;
    kp[1] = ke * s + ko * c;
  }
  __syncthreads();

  // ---- scores[h,l] = (q[h] . k[l,h]) / 16, masked ----
  for (int idx = t; idx < HH * LL; idx += 256) {
    int h = idx >> 6;
    int l = idx & 63;
    float sc = -INFINITY;
    if (sSpec[l]) {
      const float* kk = &sK[l * LDST + h * DD];
      const float* qq = &sQ[h * DD];
      float acc = 0.f;
#pragma unroll
      for (int d = 0; d < DD; ++d) acc = fmaf(qq[d], kk[d], acc);
      sc = acc * 0.0625f;
    }
    sS[h * LL + l] = sc;
  }
  __syncthreads();

  // ---- per-head masked softmax (attn = 0 if no l specified) ----
  if (t < HH) {
    float* row = &sS[t * LL];
    float m = -INFINITY;
    for (int l = 0; l < LL; ++l) m = fmaxf(m, row[l]);
    if (m == -INFINITY) {
      for (int l = 0; l < LL; ++l) row[l] = 0.f;   // -> gbar = 0 -> out = 0
    } else {
      float sum = 0.f;
      for (int l = 0; l < LL; ++l) { float e = __expf(row[l] - m); row[l] = e; sum += e; }
      float inv = 1.f / sum;
      for (int l = 0; l < LL; ++l) row[l] *= inv;
    }
  }
  __syncthreads();

  // ---- gbar[h, i] = sum_l attn[h,l] * G[l,i]  (out = attn@(G Wv^T) = (attn G) Wv^T) ----
  {
    float acc[HH];
#pragma unroll
    for (int h = 0; h < HH; ++h) acc[h] = 0.f;
    for (int l = 0; l < LL; ++l) {
      float g = sG[l * LDST + t];          // conflict-free column read
#pragma unroll
      for (int h = 0; h < HH; ++h) acc[h] = fmaf(sS[h * LL + l], g, acc[h]);
    }
#pragma unroll
    for (int h = 0; h < HH; ++h) sGB[h * MM + t] = acc[h];
  }
  __syncthreads();

  // ---- out[o] = gbar[h(o), :] . Wv[o, :] ----
  {
    const int o = t;
    const int h = t >> 5;
    const float* wv = value_weight + (size_t)o * MM;   // L2-hot
    const float* gb = &sGB[h * MM];                    // broadcast LDS reads per wave
    float acc = 0.f;
    for (int i = 0; i < MM; i += 4) {
      float4 w = *(const float4*)(wv + i);
      acc = fmaf(gb[i + 0], w.x, acc);
      acc = fmaf(gb[i + 1], w.y, acc);
      acc = fmaf(gb[i + 2], w.z, acc);
      acc = fmaf(gb[i + 3], w.w, acc);
    }
    out[(size_t)n * MM + o] = acc;
  }
}

extern "C" void kernel_launch(void* const* d_in, const int* in_sizes, int n_in,
                              void* d_out, int out_size, void* d_ws, size_t ws_size,
                              hipStream_t stream) {
  const float* sparse_values = (const float*)d_in[0];
  const int*   linear_index  = (const int*)  d_in[1];
  const int*   is_specified  = (const int*)  d_in[2];
  const float* query         = (const float*)d_in[3];
  const float* key_weight    = (const float*)d_in[4];
  const float* value_weight  = (const float*)d_in[5];
  const float* key_positions = (const float*)d_in[6];
  const float* rope_freqs    = (const float*)d_in[7];
  // d_in[8] = n_heads scalar (fixed H=8 here)

  float* out      = (float*)d_out;
  float* out_spec = out + (size_t)NN * MM;   // second tuple element, flattened

  bsisa_kernel<<<NN, 256, 0, stream>>>(sparse_values, linear_index, is_specified,
                                       query, key_weight, value_weight,
                                       key_positions, rope_freqs, out, out_spec);
}